// MIND_55344948576558
// MI455X (gfx1250) — compile-verified
//
#include <hip/hip_runtime.h>
#include <hip/hip_bf16.h>
#include <math.h>

typedef float v2f __attribute__((ext_vector_type(2)));
typedef float v8f __attribute__((ext_vector_type(8)));
typedef int   v4i __attribute__((ext_vector_type(4)));

constexpr int kBS = 4096;
constexpr int kL  = 200;
constexpr int kLP = 208;      // padded to 13 * 16
constexpr int kD  = 64;
constexpr int kK  = 4;
constexpr int kF  = 256;
constexpr int kRT = 13;       // row tiles of 16 covering kLP
#define DROPF (-2147483648.0f)

#define AS1 __attribute__((address_space(1)))
#define AS3 __attribute__((address_space(3)))

#if __has_builtin(__builtin_amdgcn_global_load_async_to_lds_b128) && \
    __has_builtin(__builtin_amdgcn_global_load_async_to_lds_b32)
#define HAVE_ASYNC_LDS 1
#else
#define HAVE_ASYNC_LDS 0
#endif

__device__ __forceinline__ void wait_async_all() {
#if HAVE_ASYNC_LDS
#if __has_builtin(__builtin_amdgcn_s_wait_asynccnt)
    __builtin_amdgcn_s_wait_asynccnt(0);
#else
    asm volatile("s_wait_asynccnt 0" ::: "memory");
#endif
#endif
}

__device__ __forceinline__ float wsum(float v) {
#pragma unroll
    for (int m = 16; m >= 1; m >>= 1) v += __shfl_xor(v, m, 32);
    return v;
}
__device__ __forceinline__ float wmaxr(float v) {
#pragma unroll
    for (int m = 16; m >= 1; m >>= 1) v = fmaxf(v, __shfl_xor(v, m, 32));
    return v;
}

__global__ __launch_bounds__(256) void mind_fused(
    const int* __restrict__ his, const float* __restrict__ E,
    const float* __restrict__ S, const float* __restrict__ B0,
    const float* __restrict__ W1, const float* __restrict__ b1,
    const float* __restrict__ W2, const float* __restrict__ b2,
    float* __restrict__ out)
{
    extern __shared__ char smem[];
    float* sP   = (float*)smem;          // kLP * kD  : hisP
    float* sS   = sP  + kLP * kD;        // kD * kD   : S
    float* sB   = sS  + kD * kD;         // kK * kLP  : routing logits
    float* sWt  = sB  + kK * kLP;        // kK * kLP  : softmax weights
    float* sC   = sWt + kK * kLP;        // kK * kD   : capsules
    float* sH   = sC  + kK * kD;         // kK * kF   : MLP hidden
    int*   sHis = (int*)(sH + kK * kF);  // kLP       : item ids (0 = pad)

    const int b    = blockIdx.x;
    const int tid  = threadIdx.x;
    const int wv   = tid >> 5;
    const int lane = tid & 31;
    const int half = lane >> 4;   // 0: lanes 0-15, 1: lanes 16-31
    const int ln   = lane & 15;

    // ---------------- phase 0: stage S, his, B0 ----------------
#if HAVE_ASYNC_LDS
    {
        // S: 4096 floats = 1024 x b128, direct global -> LDS (ASYNCcnt path)
        for (int i = tid * 4; i < kD * kD; i += 256 * 4)
            __builtin_amdgcn_global_load_async_to_lds_b128(
                (AS1 v4i*)(S + i), (AS3 v4i*)(sS + i), 0, 0);
        if (tid < kL) {
            // B0: each of 200 lanes moves one b128 chunk into the padded sB rows
            const int k = tid / 50, j = (tid % 50) * 4;
            __builtin_amdgcn_global_load_async_to_lds_b128(
                (AS1 v4i*)(B0 + k * kL + j), (AS3 v4i*)(sB + k * kLP + j), 0, 0);
            // his ids
            __builtin_amdgcn_global_load_async_to_lds_b32(
                (AS1 int*)(his + b * kL + tid), (AS3 int*)(sHis + tid), 0, 0);
        }
        // zero the L->LP padding (plain DS stores)
        if (tid < kLP - kL) sHis[kL + tid] = 0;
        if (tid < kK * (kLP - kL)) {
            const int k = tid / (kLP - kL), j = tid % (kLP - kL);
            sB[k * kLP + kL + j] = 0.0f;
        }
    }
#else
    for (int i = tid; i < kD * kD; i += 256) sS[i] = S[i];
    for (int i = tid; i < kLP; i += 256) sHis[i] = (i < kL) ? his[b * kL + i] : 0;
    for (int i = tid; i < kK * kLP; i += 256) {
        int k = i / kLP, l = i - k * kLP;
        sB[i] = (l < kL) ? B0[k * kL + l] : 0.0f;
    }
#endif
    // warm L2 for the MLP weights (shared by all 4096 workgroups)
    for (int i = tid * 32; i < kD * kF; i += 256 * 32) __builtin_prefetch(W1 + i, 0, 1);
    for (int i = tid * 32; i < kF * kD; i += 256 * 32) __builtin_prefetch(W2 + i, 0, 1);
    wait_async_all();
    __syncthreads();

    // ------- phase 1: hisP = gather(E, his) @ S  (f32 WMMA 16x16x4) -------
    for (int rt = wv; rt < kRT; rt += 8) {
        const int m  = rt * 16 + ln;
        const int id = sHis[m];                       // 0 => zero row (padding_idx)
        const float* row = E + (size_t)id * kD;
        float a[32];
#pragma unroll
        for (int kk = 0; kk < 16; ++kk) {
            if (id != 0) {
                float2 v = *(const float2*)(row + 4 * kk + 2 * half);
                a[2 * kk] = v.x; a[2 * kk + 1] = v.y;
            } else {
                a[2 * kk] = 0.0f; a[2 * kk + 1] = 0.0f;
            }
        }
#pragma unroll
        for (int ct = 0; ct < 4; ++ct) {
            v8f c = {};
#pragma unroll
            for (int kk = 0; kk < 16; ++kk) {
                const int r0 = 4 * kk + 2 * half;
                v2f av; av.x = a[2 * kk]; av.y = a[2 * kk + 1];
                v2f bv; bv.x = sS[r0 * kD + ct * 16 + ln];
                        bv.y = sS[(r0 + 1) * kD + ct * 16 + ln];
                c = __builtin_amdgcn_wmma_f32_16x16x4_f32(false, av, false, bv,
                                                          (short)0, c, false, false);
            }
#pragma unroll
            for (int v = 0; v < 8; ++v)
                sP[(rt * 16 + v + 8 * half) * kD + ct * 16 + ln] = c[v];
        }
    }
    __syncthreads();

    // ---------------- phase 2: dynamic routing (R = 3) ----------------
    for (int r = 0; r < 3; ++r) {
        // masked softmax over L, one wave per capsule k
        if (wv < kK) {
            const int k = wv;
            float mx = -3.0e38f;
            for (int l = lane; l < kL; l += 32) {
                float lg = (sHis[l] != 0) ? sB[k * kLP + l] : DROPF;
                mx = fmaxf(mx, lg);
            }
            mx = wmaxr(mx);
            float s = 0.0f;
            for (int l = lane; l < kL; l += 32) {
                float lg = (sHis[l] != 0) ? sB[k * kLP + l] : DROPF;
                float e  = expf(lg - mx);
                sWt[k * kLP + l] = e; s += e;
            }
            s = wsum(s);
            const float inv = 1.0f / s;
            for (int l = lane; l < kL; l += 32) sWt[k * kLP + l] *= inv;
            if (lane < (kLP - kL)) sWt[k * kLP + kL + lane] = 0.0f;   // zero K-pad
        }
        __syncthreads();

        // caps = W @ hisP : (16pad x 208) * (208 x 64), waves 0..3 take N-tiles
        if (wv < 4) {
            const int ct = wv;
            v8f c = {};
#pragma unroll 4
            for (int st = 0; st < kLP / 4; ++st) {
                const int r0 = 4 * st + 2 * half;
                v2f av = {0.0f, 0.0f};
                if (ln < kK) {
                    float2 t = *(const float2*)(sWt + ln * kLP + r0);
                    av.x = t.x; av.y = t.y;
                }
                v2f bv; bv.x = sP[r0 * kD + ct * 16 + ln];
                        bv.y = sP[(r0 + 1) * kD + ct * 16 + ln];
                c = __builtin_amdgcn_wmma_f32_16x16x4_f32(false, av, false, bv,
                                                          (short)0, c, false, false);
            }
#pragma unroll
            for (int v = 0; v < 8; ++v) {
                const int M = v + 8 * half;
                if (M < kK) sC[M * kD + ct * 16 + ln] = c[v];
            }
        }
        __syncthreads();

        // squash, one wave per capsule
        if (wv < kK) {
            const int k = wv;
            float x0 = sC[k * kD + lane], x1 = sC[k * kD + 32 + lane];
            float n2 = wsum(x0 * x0 + x1 * x1);
            float sc = n2 / ((1.0f + n2) * sqrtf(n2) + 1e-9f);
            sC[k * kD + lane]      = x0 * sc;
            sC[k * kD + 32 + lane] = x1 * sc;
        }
        __syncthreads();

        if (r < 2) {
            // B += caps @ hisP^T : (16pad x 64) * (64 x 208), 13 N-tiles / 8 waves
            for (int nt = wv; nt < kRT; nt += 8) {
                const int lcol = nt * 16 + ln;
                v8f c = {};
#pragma unroll
                for (int st = 0; st < 16; ++st) {
                    const int r0 = 4 * st + 2 * half;
                    v2f av = {0.0f, 0.0f};
                    if (ln < kK) {
                        float2 t = *(const float2*)(sC + ln * kD + r0);
                        av.x = t.x; av.y = t.y;
                    }
                    float2 tb = *(const float2*)(sP + lcol * kD + r0);
                    v2f bv; bv.x = tb.x; bv.y = tb.y;
                    c = __builtin_amdgcn_wmma_f32_16x16x4_f32(false, av, false, bv,
                                                              (short)0, c, false, false);
                }
#pragma unroll
                for (int v = 0; v < 8; ++v) {
                    const int M = v + 8 * half;
                    if (M < kK) sB[M * kLP + lcol] += c[v];   // pad rows of sP are 0
                }
            }
            __syncthreads();
        }
    }

    // ---------------- phase 3: MLP ----------------
    // h = relu(caps @ W1 + b1) : (16pad x 64) * (64 x 256), 16 N-tiles / 8 waves
    for (int nt = wv; nt < 16; nt += 8) {
        v8f c = {};
#pragma unroll
        for (int st = 0; st < 16; ++st) {
            const int r0 = 4 * st + 2 * half;
            v2f av = {0.0f, 0.0f};
            if (ln < kK) {
                float2 t = *(const float2*)(sC + ln * kD + r0);
                av.x = t.x; av.y = t.y;
            }
            v2f bv; bv.x = W1[r0 * kF + nt * 16 + ln];
                    bv.y = W1[(r0 + 1) * kF + nt * 16 + ln];
            c = __builtin_amdgcn_wmma_f32_16x16x4_f32(false, av, false, bv,
                                                      (short)0, c, false, false);
        }
        const float bias = b1[nt * 16 + ln];
#pragma unroll
        for (int v = 0; v < 8; ++v) {
            const int M = v + 8 * half;
            if (M < kK) sH[M * kF + nt * 16 + ln] = fmaxf(c[v] + bias, 0.0f);
        }
    }
    __syncthreads();

    // out = h @ W2 + b2 : (16pad x 256) * (256 x 64), waves 0..3 take N-tiles
    if (wv < 4) {
        const int ct = wv;
        v8f c = {};
#pragma unroll 4
        for (int st = 0; st < kF / 4; ++st) {
            const int r0 = 4 * st + 2 * half;
            v2f av = {0.0f, 0.0f};
            if (ln < kK) {
                float2 t = *(const float2*)(sH + ln * kF + r0);
                av.x = t.x; av.y = t.y;
            }
            v2f bv; bv.x = W2[r0 * kD + ct * 16 + ln];
                    bv.y = W2[(r0 + 1) * kD + ct * 16 + ln];
            c = __builtin_amdgcn_wmma_f32_16x16x4_f32(false, av, false, bv,
                                                      (short)0, c, false, false);
        }
        const float bias = b2[ct * 16 + ln];
#pragma unroll
        for (int v = 0; v < 8; ++v) {
            const int M = v + 8 * half;
            if (M < kK) out[(size_t)b * kK * kD + M * kD + ct * 16 + ln] = c[v] + bias;
        }
    }
}

extern "C" void kernel_launch(void* const* d_in, const int* in_sizes, int n_in,
                              void* d_out, int out_size, void* d_ws, size_t ws_size,
                              hipStream_t stream) {
    const int*   his = (const int*)d_in[0];
    const float* E   = (const float*)d_in[1];
    const float* S   = (const float*)d_in[2];
    const float* B0  = (const float*)d_in[3];
    const float* W1  = (const float*)d_in[4];
    const float* b1  = (const float*)d_in[5];
    const float* W2  = (const float*)d_in[6];
    const float* b2  = (const float*)d_in[7];
    float* out = (float*)d_out;

    const size_t smem =
        (size_t)(kLP * kD + kD * kD + 2 * kK * kLP + kK * kD + kK * kF) * sizeof(float)
        + (size_t)kLP * sizeof(int);   // = 82,240 bytes

    mind_fused<<<dim3(kBS), dim3(256), smem, stream>>>(his, E, S, B0, W1, b1, W2, b2, out);
}